// RadonLayer_88691074662809
// MI455X (gfx1250) — compile-verified
//
#include <hip/hip_runtime.h>

// Problem constants (from the reference):
//   input  x: (4, 1, 384, 384) f32
//   pad = 38 -> padded 460x460, N_ANGLES = 64
//   output: (4, 460, 64) f32, out[b,y,a] = (1/460) * sum_x bilerp(line(a,y,x))
#define IW      384
#define PAD     38
#define HP      460          // padded H == padded W
#define NA      64
#define NB      4
#define CHUNKS  115          // 460 / 4 x-samples per WMMA step

typedef __attribute__((ext_vector_type(2))) float v2f;
typedef __attribute__((ext_vector_type(8))) float v8f;

// Sample one neighbor from the UNPADDED image using padded-space integer
// coords. Zero outside the interior [PAD, PAD+IW). Branchless (clamp+mask)
// so EXEC stays all-ones for the WMMA ops.
__device__ __forceinline__ float fetch_img(const float* __restrict__ img,
                                           int py, int px) {
  unsigned ux = (unsigned)(px - PAD);
  unsigned uy = (unsigned)(py - PAD);
  bool ok = (ux < (unsigned)IW) && (uy < (unsigned)IW);
  int idx = ok ? (int)(uy * IW + ux) : 0;
  float v = img[idx];
  return ok ? v : 0.0f;
}

__device__ __forceinline__ float bsample(const float* __restrict__ img,
                                         float fx, float fy) {
  float fx0 = floorf(fx), fy0 = floorf(fy);
  float wx = fx - fx0,    wy = fy - fy0;
  int x0 = (int)fx0, y0 = (int)fy0;
  float v00 = fetch_img(img, y0,     x0);
  float v01 = fetch_img(img, y0,     x0 + 1);
  float v10 = fetch_img(img, y0 + 1, x0);
  float v11 = fetch_img(img, y0 + 1, x0 + 1);
  float top = fmaf(wx, v01 - v00, v00);
  float bot = fmaf(wx, v11 - v10, v10);
  return fmaf(wy, bot - top, top);
}

// One wave32 per 16-row output tile of one (batch, angle).
// Per iteration the wave gathers a 16(y) x 4(x) f32 sample block laid out as
// the WMMA 32-bit A-matrix (lanes 0-15: K={0,1}, lanes 16-31: K={2,3}) and
// reduces it with V_WMMA_F32_16X16X4_F32 against a ones-column B.
__global__ __launch_bounds__(256) void radon_wmma_kernel(
    const float* __restrict__ x, float* __restrict__ out) {
  const int bid  = blockIdx.x;
  const int grp  = bid & 3;          // 4 tile-groups of 8 waves -> 32 tiles >= 29
  const int ba   = bid >> 2;
  const int ang  = ba & (NA - 1);
  const int b    = ba >> 6;
  const int wave = threadIdx.x >> 5;
  const int lane = threadIdx.x & 31;
  const int tile = grp * 8 + wave;
  const int yt0  = tile * 16;
  if (yt0 >= HP) return;             // wave-uniform exit

  const float* __restrict__ img = x + (size_t)b * IW * IW;

  // Warm the L2 path for this image (gfx1250 global_prefetch_b8).
  __builtin_prefetch(img + (size_t)threadIdx.x * 64, 0, 0);

  // theta = ang * (pi/64); line param in padded coords (Hp == Wp == 460):
  //   ix = c*(x - 229.5) - s*(y - 229.5) + 229.5
  //   iy = s*(x - 229.5) + c*(y - 229.5) + 229.5
  const float th  = (float)ang * (3.14159265358979323846f / 64.0f);
  const float cth = cosf(th);
  const float sth = sinf(th);

  const int   m     = lane & 15;           // output row within tile (A-matrix M)
  const int   kbase = (lane >> 4) << 1;    // x offset within 4-chunk (A-matrix K)
  const float yf    = (float)(yt0 + m) - 229.5f;
  const float bx    = fmaf(-sth, yf, 229.5f);
  const float by    = fmaf( cth, yf, 229.5f);

  v8f acc = {};

#if __has_builtin(__builtin_amdgcn_wmma_f32_16x16x4_f32)
  // B (4x16): column 0 = ones. Row-striped across lanes -> B[k][0] lives in
  // lanes 0 and 16 of both B VGPRs.
  const float bl = ((lane & 15) == 0) ? 1.0f : 0.0f;
  v2f Bm; Bm.x = bl; Bm.y = bl;

  for (int ch = 0; ch < CHUNKS; ++ch) {
    const float xf0 = (float)(ch * 4 + kbase) - 229.5f;
    const float xf1 = xf0 + 1.0f;
    // 8 independent gathers per lane -> deep memory-level parallelism;
    // address math co-executes with the previous WMMA (XDL ~ TRANS class).
    const float s0 = bsample(img, fmaf(cth, xf0, bx), fmaf(sth, xf0, by));
    const float s1 = bsample(img, fmaf(cth, xf1, bx), fmaf(sth, xf1, by));
    v2f A; A.x = s0; A.y = s1;
    acc = __builtin_amdgcn_wmma_f32_16x16x4_f32(
        /*neg_a=*/false, A, /*neg_b=*/false, Bm,
        /*c_mod=*/(short)0, acc, /*reuse_a=*/false, /*reuse_b=*/false);
  }

  // Row sums live in column N==0: lane 0 -> M=0..7 (acc[0..7]),
  // lane 16 -> M=8..15 (acc[0..7]).
  if ((lane & 15) == 0) {
    const int mb = (lane >> 4) * 8;
    #pragma unroll
    for (int r = 0; r < 8; ++r) {
      const int y = yt0 + mb + r;
      if (y < HP)
        out[((size_t)b * HP + y) * NA + ang] = acc[r] * (1.0f / 460.0f);
    }
  }
#else
  // Fallback (no WMMA builtin): per-lane partials + cross-half shuffle reduce.
  float partial = 0.0f;
  for (int ch = 0; ch < CHUNKS; ++ch) {
    const float xf0 = (float)(ch * 4 + kbase) - 229.5f;
    const float xf1 = xf0 + 1.0f;
    partial += bsample(img, fmaf(cth, xf0, bx), fmaf(sth, xf0, by));
    partial += bsample(img, fmaf(cth, xf1, bx), fmaf(sth, xf1, by));
  }
  partial += __shfl_xor(partial, 16, 32);
  if (lane < 16) {
    const int y = yt0 + m;
    if (y < HP)
      out[((size_t)b * HP + y) * NA + ang] = partial * (1.0f / 460.0f);
  }
#endif
}

extern "C" void kernel_launch(void* const* d_in, const int* in_sizes, int n_in,
                              void* d_out, int out_size, void* d_ws, size_t ws_size,
                              hipStream_t stream) {
  (void)in_sizes; (void)n_in; (void)out_size; (void)d_ws; (void)ws_size;
  const float* x = (const float*)d_in[0];
  float* out = (float*)d_out;
  // 4 batches * 64 angles * 4 tile-groups; 256 threads = 8 waves/block,
  // each wave owns one 16-row output tile.
  dim3 grid(NB * NA * 4);
  radon_wmma_kernel<<<grid, dim3(256), 0, stream>>>(x, out);
}